// TritonChunkedKasminaLayer_40200893890919
// MI455X (gfx1250) — compile-verified
//
#include <hip/hip_runtime.h>

// Problem constants from the reference (shapes fixed by setup_inputs()).
#define HDIM   8192          // H: features per row
#define NCHUNK 1000          // CHUNKS
#define NBP    10            // NUM_BP
#define ACTIVE_LO 3
#define ACTIVE_HI 6

typedef float v4f __attribute__((ext_vector_type(4)));
typedef int   v4i __attribute__((ext_vector_type(4)));

#define GLOBAL_AS __attribute__((address_space(1)))
#define LDS_AS    __attribute__((address_space(3)))

#if __has_builtin(__builtin_amdgcn_global_load_async_to_lds_b128)
#define HAVE_ASYNC_LDS 1
#else
#define HAVE_ASYNC_LDS 0
#endif

#if __has_builtin(__builtin_amdgcn_s_wait_asynccnt)
#define WAIT_ASYNC(n) __builtin_amdgcn_s_wait_asynccnt(n)
#else
#define WAIT_ASYNC(n) asm volatile("s_wait_asynccnt %0" :: "i"(n) : "memory")
#endif

// Each thread owns 4 consecutive columns (one b128 lane-quad).
// blockDim.x = 256 -> 1024 columns per block -> gridDim.x = 8.
// gridDim.y blocks stride over rows.
__global__ __launch_bounds__(256) void kasmina_affine_stream(
    const float* __restrict__ x,
    const float* __restrict__ bw,     // [NBP, HDIM]
    const int*   __restrict__ ls,     // [NCHUNK]
    const int*   __restrict__ bids,   // [NCHUNK]
    const int*   __restrict__ gs,     // [NCHUNK]
    float*       __restrict__ out,
    int nrows)
{
    const int tid  = threadIdx.x;
    const int col0 = (blockIdx.x * blockDim.x + tid) * 4;

    // ---- per-column preamble: fold all control flow into (m, a) ----------
    // col0 % 8 is 0 or 4, so all 4 columns share one seed chunk -> load the
    // metadata exactly once per thread.
    const int sid = col0 >> 3;
    int mode = -1;                      // -1 = passthrough
    int bc   = 0;
    if (sid < NCHUNK) {
        const int state = ls[sid];
        const int bid   = bids[sid];
        const int strat = gs[sid];
        const bool active = (state >= ACTIVE_LO) & (state <= ACTIVE_HI) &
                            (bid < NBP);
        if (active) {
            mode = strat;
            bc   = bid < 0 ? 0 : (bid > NBP - 1 ? NBP - 1 : bid);
        }
    }
    // One b128 load for the 4 gathered weights (bc=0 when inactive: safe).
    const v4f w4 = *(const v4f*)(bw + (size_t)bc * HDIM + col0);

    v4f m, a;
#pragma unroll
    for (int k = 0; k < 4; ++k) {
        float mm = 1.0f, aa = 0.0f;          // passthrough default
        if (mode == 0)      { mm = w4[k];  aa = 0.0f;         } // x * w
        else if (mode == 1) { mm = 1.0f;   aa = w4[k];        } // x + w
        else if (mode >= 2) { mm = 0.5f;   aa = 0.5f * w4[k]; } // blend
        m[k] = mm;
        a[k] = aa;
    }

    const size_t rowStride = (size_t)gridDim.y * HDIM;
    size_t idx = (size_t)blockIdx.y * HDIM + col0;

#if HAVE_ASYNC_LDS
    // ---- double-buffered async stream: HBM -> LDS (ASYNCcnt) -> VGPR -----
    // Each lane async-loads exactly the 16B it will consume, so a per-wave
    // s_wait_asynccnt is the only synchronization needed (no barriers).
    __shared__ v4f lbuf[2][256];

    if ((int)blockIdx.y < nrows) {
        __builtin_amdgcn_global_load_async_to_lds_b128(
            (GLOBAL_AS v4i*)(x + idx),
            (LDS_AS v4i*)&lbuf[0][tid], 0, 0);
    }
    int sel = 0;
    for (int row = blockIdx.y; row < nrows; row += gridDim.y, idx += rowStride) {
        const bool hasNext = (row + (int)gridDim.y) < nrows;
        if (hasNext) {
            __builtin_amdgcn_global_load_async_to_lds_b128(
                (GLOBAL_AS v4i*)(x + idx + rowStride),
                (LDS_AS v4i*)&lbuf[sel ^ 1][tid], 0, 0);
            WAIT_ASYNC(1);      // oldest (current buffer) complete, next in flight
        } else {
            WAIT_ASYNC(0);      // drain: current buffer complete
        }
        const v4f xv = lbuf[sel][tid];       // ds_load_b128
        v4f r;
        r.x = __builtin_fmaf(m.x, xv.x, a.x);
        r.y = __builtin_fmaf(m.y, xv.y, a.y);
        r.z = __builtin_fmaf(m.z, xv.z, a.z);
        r.w = __builtin_fmaf(m.w, xv.w, a.w);
        __builtin_nontemporal_store(r, (v4f*)(out + idx));
        sel ^= 1;
    }
#else
    // ---- fallback: direct NT b128 stream with gfx1250 prefetch -----------
    for (int row = blockIdx.y; row < nrows; row += gridDim.y, idx += rowStride) {
        if (row + (int)gridDim.y < nrows) {
            __builtin_prefetch(x + idx + rowStride, 0, 1);   // global_prefetch_b8
        }
        v4f xv = __builtin_nontemporal_load((const v4f*)(x + idx));
        v4f r;
        r.x = __builtin_fmaf(m.x, xv.x, a.x);
        r.y = __builtin_fmaf(m.y, xv.y, a.y);
        r.z = __builtin_fmaf(m.z, xv.z, a.z);
        r.w = __builtin_fmaf(m.w, xv.w, a.w);
        __builtin_nontemporal_store(r, (v4f*)(out + idx));
    }
#endif
}

extern "C" void kernel_launch(void* const* d_in, const int* in_sizes, int n_in,
                              void* d_out, int out_size, void* d_ws, size_t ws_size,
                              hipStream_t stream) {
    (void)in_sizes; (void)n_in; (void)d_ws; (void)ws_size;
    const float* x    = (const float*)d_in[0];
    const float* bw   = (const float*)d_in[1];
    const int*   ls   = (const int*)d_in[2];
    const int*   bids = (const int*)d_in[3];
    const int*   gs   = (const int*)d_in[4];
    float*       out  = (float*)d_out;

    const int nrows = out_size / HDIM;          // 8192 for the reference shapes

    dim3 block(256, 1, 1);
    int gy = nrows < 1024 ? (nrows > 0 ? nrows : 1) : 1024;
    dim3 grid(HDIM / (256 * 4), gy, 1);

    kasmina_affine_stream<<<grid, block, 0, stream>>>(x, bw, ls, bids, gs, out, nrows);
}